// CustomDOMINANT_7559142441736
// MI455X (gfx1250) — compile-verified
//
#include <hip/hip_runtime.h>

typedef __attribute__((ext_vector_type(2))) float v2f;
typedef __attribute__((ext_vector_type(8))) float v8f;

#define IN_DIM 256
#define HID 64

// ---------------------------------------------------------------------------
// Elementwise / prep kernels
// ---------------------------------------------------------------------------
__global__ void k_fill_f32(float* __restrict__ p, float v, long n) {
    long i = (long)blockIdx.x * blockDim.x + threadIdx.x;
    if (i < n) p[i] = v;
}

// deg[dst] += 1 for each of the E raw edges (self loops pre-counted by init=1)
__global__ void k_deg_accum(const int* __restrict__ ei, float* __restrict__ deg, int E) {
    int e = blockIdx.x * blockDim.x + threadIdx.x;
    if (e < E) atomicAdd(&deg[ei[E + e]], 1.0f);
}

__global__ void k_rsqrt(const float* __restrict__ deg, float* __restrict__ dinv, int n) {
    int i = blockIdx.x * blockDim.x + threadIdx.x;
    if (i < n) {
        float d = deg[i];
        dinv[i] = d > 0.0f ? rsqrtf(d) : 0.0f;
    }
}

// out[i,f] = bias[f] + t[i,f] * dinv[i]^2   (bias + self-loop term, no atomics)
__global__ void k_conv_init(const float* __restrict__ t, const float* __restrict__ bias,
                            const float* __restrict__ dinv, float* __restrict__ out,
                            int n, int dim) {
    long g = (long)blockIdx.x * blockDim.x + threadIdx.x;
    if (g >= (long)n * dim) return;
    int i = (int)(g / dim);
    int f = (int)(g % dim);
    float di = dinv[i];
    out[g] = bias[f] + t[g] * di * di;
}

// out[dst] += t[src] * (dinv[src]*dinv[dst]) over E edges; 4 floats per thread.
// Destinations are L2-resident (<=12.6 MB), so atomics resolve at L2 atomic units.
__global__ void k_scatter_edges(const float* __restrict__ t, const int* __restrict__ ei,
                                const float* __restrict__ dinv, float* __restrict__ out,
                                int E, int dim) {
    long g = (long)blockIdx.x * blockDim.x + threadIdx.x;
    int chunks = dim >> 2;
    long tot = (long)E * chunks;
    if (g >= tot) return;
    int e  = (int)(g / chunks);
    int f4 = (int)(g % chunks);
    int s = ei[e];
    int d = ei[E + e];
    float nrm = dinv[s] * dinv[d];
    const float4 v = *(const float4*)(t + (size_t)s * dim + (size_t)f4 * 4);
    float* o = out + (size_t)d * dim + (size_t)f4 * 4;
    atomicAdd(o + 0, v.x * nrm);
    atomicAdd(o + 1, v.y * nrm);
    atomicAdd(o + 2, v.z * nrm);
    atomicAdd(o + 3, v.w * nrm);
}

__global__ void k_relu(float* __restrict__ p, long n) {
    long i = (long)blockIdx.x * blockDim.x + threadIdx.x;
    if (i < n) p[i] = fmaxf(p[i], 0.0f);
}

// x_cf = x with column 0 flipped to (1 - x[:,0])
__global__ void k_make_xcf(const float* __restrict__ x, float* __restrict__ xcf, long n) {
    long g = (long)blockIdx.x * blockDim.x + threadIdx.x;
    if (g >= n) return;
    float v = x[g];
    if ((g % IN_DIM) == 0) v = 1.0f - v;
    xcf[g] = v;
}

// z (N x 64) -> z_s (N x 32), z_ns (N x 32)
__global__ void k_split_z(const float* __restrict__ z, float* __restrict__ zs,
                          float* __restrict__ zns, int n) {
    long g = (long)blockIdx.x * blockDim.x + threadIdx.x;
    if (g >= (long)n * 32) return;
    int i = (int)(g / 32);
    int f = (int)(g % 32);
    zs[g]  = z[(size_t)i * HID + f];
    zns[g] = z[(size_t)i * HID + 32 + f];
}

__global__ void k_split_s_only(const float* __restrict__ zcf, float* __restrict__ zscf, int n) {
    long g = (long)blockIdx.x * blockDim.x + threadIdx.x;
    if (g >= (long)n * 32) return;
    int i = (int)(g / 32);
    int f = (int)(g % 32);
    zscf[g] = zcf[(size_t)i * HID + f];
}

// ---------------------------------------------------------------------------
// FP32 WMMA GEMM: C[M,Nc] = A[M,K] @ W[K,Nc], row-major, all dims %16 == 0.
// One wave -> one 16x16 tile; K accumulated 4 at a time via V_WMMA_F32_16X16X4_F32.
// ---------------------------------------------------------------------------
__global__ void k_gemm_wmma_nn(const float* __restrict__ A, const float* __restrict__ W,
                               float* __restrict__ C, int M, int K, int Nc) {
    const int lane = threadIdx.x & 31;
    const int wave = threadIdx.x >> 5;
    const int tile = blockIdx.x * (blockDim.x >> 5) + wave;
    const int ntj = Nc >> 4;
    const int ntiles = (M >> 4) * ntj;
    if (tile >= ntiles) return;               // uniform per wave
    const int ti = tile / ntj;
    const int tj = tile % ntj;
    const int half = lane >> 4;
    const int idx  = lane & 15;

    const float* arow = A + (size_t)(ti * 16 + idx) * K;      // A row m = 16*ti+idx
    const float* wcol = W + (tj * 16 + idx);                  // B col n = 16*tj+idx

    v8f acc = {};
    for (int k0 = 0; k0 < K; k0 += 4) {
        const int ko = k0 + 2 * half;
        v2f a = *(const v2f*)(arow + ko);                     // contiguous -> b64
        v2f b;
        b.x = wcol[(size_t)(ko + 0) * Nc];
        b.y = wcol[(size_t)(ko + 1) * Nc];
        acc = __builtin_amdgcn_wmma_f32_16x16x4_f32(false, a, false, b,
                                                    (short)0, acc, false, false);
    }
    float* crow = C + (size_t)(ti * 16 + 8 * half) * Nc + (size_t)(tj * 16 + idx);
#pragma unroll
    for (int v = 0; v < 8; ++v) crow[(size_t)v * Nc] = acc[v];
}

// ---------------------------------------------------------------------------
// S = Hs @ Hs^T, Hs is (n x 64) fp32, n % 64 == 0. Hs (3 MB) lives in L2.
// Register-blocked 1x4: each wave computes a 16x64 strip, reusing the A
// fragment across 4 WMMAs per K-step. Output stored non-temporally (604 MB
// stream must not evict L2-resident Hs).
// ---------------------------------------------------------------------------
__global__ void k_gemm_wmma_nt64(const float* __restrict__ Hs, float* __restrict__ S, int n) {
    const int lane = threadIdx.x & 31;
    const int wave = threadIdx.x >> 5;
    const int job  = blockIdx.x * (blockDim.x >> 5) + wave;
    const int nt   = n >> 4;          // # 16-wide tiles per dim
    const int ngj  = nt >> 2;         // groups of 4 column tiles
    if (job >= nt * ngj) return;      // uniform per wave
    const int ti  = job / ngj;
    const int tjg = job % ngj;
    const int half = lane >> 4;
    const int idx  = lane & 15;

    const float* ar  = Hs + (size_t)(ti * 16 + idx) * HID;          // A row
    const float* br0 = Hs + (size_t)(tjg * 64 +  0 + idx) * HID;    // B[k][j]=Hs[j][k]
    const float* br1 = Hs + (size_t)(tjg * 64 + 16 + idx) * HID;
    const float* br2 = Hs + (size_t)(tjg * 64 + 32 + idx) * HID;
    const float* br3 = Hs + (size_t)(tjg * 64 + 48 + idx) * HID;

    v8f acc0 = {}, acc1 = {}, acc2 = {}, acc3 = {};
#pragma unroll
    for (int k0 = 0; k0 < HID; k0 += 4) {
        const int ko = k0 + 2 * half;
        v2f a  = *(const v2f*)(ar  + ko);
        v2f b0 = *(const v2f*)(br0 + ko);
        v2f b1 = *(const v2f*)(br1 + ko);
        v2f b2 = *(const v2f*)(br2 + ko);
        v2f b3 = *(const v2f*)(br3 + ko);
        acc0 = __builtin_amdgcn_wmma_f32_16x16x4_f32(false, a, false, b0, (short)0, acc0, false, false);
        acc1 = __builtin_amdgcn_wmma_f32_16x16x4_f32(false, a, false, b1, (short)0, acc1, false, false);
        acc2 = __builtin_amdgcn_wmma_f32_16x16x4_f32(false, a, false, b2, (short)0, acc2, false, false);
        acc3 = __builtin_amdgcn_wmma_f32_16x16x4_f32(false, a, false, b3, (short)0, acc3, false, false);
    }

    float* srow = S + (size_t)(ti * 16 + 8 * half) * n + (size_t)(tjg * 64 + idx);
#pragma unroll
    for (int v = 0; v < 8; ++v) {
        float* r = srow + (size_t)v * n;
        __builtin_nontemporal_store(acc0[v], r);
        __builtin_nontemporal_store(acc1[v], r + 16);
        __builtin_nontemporal_store(acc2[v], r + 32);
        __builtin_nontemporal_store(acc3[v], r + 48);
    }
}

// ---------------------------------------------------------------------------
// Host orchestration
// ---------------------------------------------------------------------------
extern "C" void kernel_launch(void* const* d_in, const int* in_sizes, int n_in,
                              void* d_out, int out_size, void* d_ws, size_t ws_size,
                              hipStream_t stream) {
    (void)n_in; (void)out_size; (void)ws_size;

    const float* x      = (const float*)d_in[0];
    const int*   ei     = (const int*)  d_in[1];
    const float* encW1  = (const float*)d_in[2];
    const float* encb1  = (const float*)d_in[3];
    const float* encW2  = (const float*)d_in[4];
    const float* encb2  = (const float*)d_in[5];
    const float* d1W1   = (const float*)d_in[6];
    const float* d1b1   = (const float*)d_in[7];
    const float* d1W2   = (const float*)d_in[8];
    const float* d1b2   = (const float*)d_in[9];
    const float* d2W1   = (const float*)d_in[10];
    const float* d2b1   = (const float*)d_in[11];
    const float* d2W2   = (const float*)d_in[12];
    const float* d2b2   = (const float*)d_in[13];
    const float* sW     = (const float*)d_in[14];
    const float* sb     = (const float*)d_in[15];

    const int N = in_sizes[0] / IN_DIM;    // 12288
    const int E = in_sizes[1] / 2;         // 393216

    // workspace carve-out (floats)
    float* w    = (float*)d_ws;
    float* deg  = w;                         // N
    float* dinv = deg  + N;                  // N
    float* xcf  = dinv + N;                  // N*256 (reused later as z_s_cf)
    float* t    = xcf  + (size_t)N * IN_DIM; // N*256 generic GEMM output
    float* hA   = t    + (size_t)N * IN_DIM; // N*64 hidden
    float* z    = hA   + (size_t)N * HID;    // N*64
    float* zcf  = z    + (size_t)N * HID;    // N*64
    float* hs   = zcf  + (size_t)N * HID;    // N*64
    float* zscf = xcf;                       // N*32 (xcf region free after cf encoder)

    // outputs
    float* o      = (float*)d_out;
    float* o_zs   = o;                                  // N*32
    float* o_zns  = o_zs   + (size_t)N * 32;            // N*32
    float* o_xs   = o_zns  + (size_t)N * 32;            // N*256
    float* o_xns  = o_xs   + (size_t)N * IN_DIM;        // N*256
    float* o_xscf = o_xns  + (size_t)N * IN_DIM;        // N*256
    float* o_s    = o_xscf + (size_t)N * IN_DIM;        // N*N

    const int TB = 256;
    auto blocks = [](long n, int tb) { return (int)((n + tb - 1) / tb); };

    // --- degrees & normalization ---
    k_fill_f32<<<blocks(N, TB), TB, 0, stream>>>(deg, 1.0f, N);  // self loops
    k_deg_accum<<<blocks(E, TB), TB, 0, stream>>>(ei, deg, E);
    k_rsqrt<<<blocks(N, TB), TB, 0, stream>>>(deg, dinv, N);

    // one full GCN conv: out = segsum((A@W)[src]*norm -> dst) + bias  (self loop folded into init)
    auto conv = [&](const float* Ain, const float* Wm, const float* bias,
                    float* outp, int K, int Nc) {
        int tiles = (N >> 4) * (Nc >> 4);
        k_gemm_wmma_nn<<<(tiles + 3) / 4, 128, 0, stream>>>(Ain, Wm, t, N, K, Nc);
        long tot = (long)N * Nc;
        k_conv_init<<<blocks(tot, TB), TB, 0, stream>>>(t, bias, dinv, outp, N, Nc);
        long st = (long)E * (Nc >> 2);
        k_scatter_edges<<<blocks(st, TB), TB, 0, stream>>>(t, ei, dinv, outp, E, Nc);
    };
    auto relu = [&](float* p, long n) {
        k_relu<<<blocks(n, TB), TB, 0, stream>>>(p, n);
    };

    // --- encoder(x) -> z ---
    conv(x, encW1, encb1, hA, IN_DIM, HID); relu(hA, (long)N * HID);
    conv(hA, encW2, encb2, z, HID, HID);

    // --- encoder(x_cf) -> zcf ---
    k_make_xcf<<<blocks((long)N * IN_DIM, TB), TB, 0, stream>>>(x, xcf, (long)N * IN_DIM);
    conv(xcf, encW1, encb1, hA, IN_DIM, HID); relu(hA, (long)N * HID);
    conv(hA, encW2, encb2, zcf, HID, HID);

    // --- split latent ---
    k_split_z<<<blocks((long)N * 32, TB), TB, 0, stream>>>(z, o_zs, o_zns, N);
    k_split_s_only<<<blocks((long)N * 32, TB), TB, 0, stream>>>(zcf, zscf, N);

    // --- dec1(z_s) -> x_s_hat ---
    conv(o_zs, d1W1, d1b1, hA, 32, HID); relu(hA, (long)N * HID);
    conv(hA, d1W2, d1b2, o_xs, HID, IN_DIM);

    // --- dec2(z_ns) -> x_ns_hat ---
    conv(o_zns, d2W1, d2b1, hA, 32, HID); relu(hA, (long)N * HID);
    conv(hA, d2W2, d2b2, o_xns, HID, IN_DIM);

    // --- dec1(z_s_cf) -> x_s_cf_hat ---
    conv(zscf, d1W1, d1b1, hA, 32, HID); relu(hA, (long)N * HID);
    conv(hA, d1W2, d1b2, o_xscf, HID, IN_DIM);

    // --- hs = conv(z_ns, s_W, s_b) ---
    conv(o_zns, sW, sb, hs, 32, HID);

    // --- s_ = hs @ hs^T (fp32 WMMA, 1x4 register-blocked, NT stores) ---
    {
        int nt   = N >> 4;            // 768
        int jobs = nt * (nt >> 2);    // 147456 wave-jobs (16x64 strips)
        k_gemm_wmma_nt64<<<(jobs + 7) / 8, 256, 0, stream>>>(hs, o_s, N);
    }
}